// DistancePenalization_7722351199040
// MI455X (gfx1250) — compile-verified
//
#include <hip/hip_runtime.h>

#define BIGF   1.0e10f
#define WANISO 0.096f
#define VOXPB  262144   // 64*64*64
#define NVOL   6        // 2 batches * 3 labels

typedef __attribute__((ext_vector_type(2))) float v2f;
typedef __attribute__((ext_vector_type(8))) float v8f;

// ---------------------------------------------------------------------------
// Wave-level f32 sum via V_WMMA_F32_16X16X4_F32 with a ones B-matrix.
// Lane L contributes s at A(M=L%16, K in {0,2}); other K slots are 0, so
// D(m,n) = s[m] + s[m+16] (rowsum). Each lane sums its 8 acc VGPRs
// (columns are identical), then an xor-16 shuffle merges the M=0..7 and
// M=8..15 half-sums -> every lane holds the wave total, full f32 precision.
// EXEC must be all ones: called only from fully-converged code.
// ---------------------------------------------------------------------------
__device__ __forceinline__ float wave_wmma_sum(float s) {
    v2f a; a.x = s;    a.y = 0.0f;
    v2f b; b.x = 1.0f; b.y = 1.0f;
    v8f c = {};
    c = __builtin_amdgcn_wmma_f32_16x16x4_f32(
            /*neg_a=*/false, a, /*neg_b=*/false, b,
            /*c_mod=*/(short)0, c, /*reuse_a=*/false, /*reuse_b=*/false);
    float t = c[0] + c[1] + c[2] + c[3] + c[4] + c[5] + c[6] + c[7];
    t += __shfl_xor(t, 16, 32);   // wave32: combine the two half-wave sums
    return t;
}

__device__ __forceinline__ float block_sum_256(float s) {
    __shared__ float wsum[8];
    float t = wave_wmma_sum(s);
    int lane = threadIdx.x & 31;
    int w    = threadIdx.x >> 5;
    if (lane == 0) wsum[w] = t;
    __syncthreads();
    float r = 0.0f;
    if (threadIdx.x == 0) {
        #pragma unroll
        for (int i = 0; i < 8; ++i) r += wsum[i];
    }
    return r;  // valid in thread 0 only
}

// ---------------------------------------------------------------------------
// One 1-D min-plus (quadratic-cost) pass over all 6 (batch,label) volumes.
// axis: 2 = z (stride 1), 1 = y (stride 64), 0 = x (stride 4096).
// labels != nullptr => fused field init f = (y==l0 ? 0 : BIG) (first pass).
// Block = 256 threads = 4 lines staged in LDS.
//   - src passes use CDNA5 GLOBAL_LOAD_ASYNC_TO_LDS_B32: memory -> LDS
//     directly (no VGPR round-trip), tracked by ASYNCcnt.
//   - each thread then produces one output via a fully unrolled 64-tap
//     fmin/fmaf sweep (same-address LDS reads broadcast, conflict-free).
// ---------------------------------------------------------------------------
__global__ void minplus_pass(const float* __restrict__ src,
                             const int*   __restrict__ labels,
                             float*       __restrict__ dst,
                             int axis) {
    __shared__ float line[4][64];
    const int lb    = threadIdx.x >> 6;          // line within block, 0..3
    const int j     = threadIdx.x & 63;          // position along line
    const int gline = (blockIdx.x << 2) + lb;    // 0 .. 6*4096-1
    const int vol   = gline >> 12;               // 0..5  (b*3 + l0)
    const int l     = gline & 4095;              // line index inside volume

    int base, stride;
    if (axis == 2)      { base = l << 6;                        stride = 1;    }
    else if (axis == 1) { base = ((l >> 6) << 12) + (l & 63);   stride = 64;   }
    else                { base = l;                             stride = 4096; }

    const size_t volOff = (size_t)vol * VOXPB;
    const int    idx    = base + j * stride;

    if (labels) {
        const int b  = vol / 3;
        const int l0 = vol - 3 * b;
        line[lb][j] =
            (labels[(size_t)b * VOXPB + idx] == l0) ? 0.0f : BIGF;
        __syncthreads();
    } else {
        // CDNA5 async global->LDS copy (GVS mode: saddr base + 32-bit voffset)
        unsigned ldsOff = (unsigned)(uintptr_t)(&line[lb][j]); // LDS byte offset
        unsigned goff   = (unsigned)((volOff + (size_t)idx) * 4u);
        asm volatile("global_load_async_to_lds_b32 %0, %1, %2"
                     :
                     : "v"(ldsOff), "v"(goff), "s"(src)
                     : "memory");
        asm volatile("s_wait_asynccnt 0" ::: "memory");
        __syncthreads();
    }

    const float* ln = line[lb];
    float m = BIGF;
    #pragma unroll
    for (int i = 0; i < 64; ++i) {
        float d = WANISO * (float)(i - j);
        m = fminf(m, fmaf(d, d, ln[i]));
    }
    dst[volOff + idx] = m;
}

// ---------------------------------------------------------------------------
// Per-voxel loss terms + deterministic two-stage reduction (WMMA wave sums).
// contribution(v) = dt * sum_{c=1,2} [ (y==c) ? (1-x_c) : x_c ]
// with dt = sqrt(min over labels l != y of D_l) + 1.
// 4 consecutive voxels per thread -> all streams load as B128.
// ---------------------------------------------------------------------------
__global__ void reduce_kernel(const float* __restrict__ x,
                              const int*   __restrict__ y,
                              const float* __restrict__ D,
                              float*       __restrict__ partials) {
    const int nvox = 2 * VOXPB;
    float acc = 0.0f;
    for (int g4 = (blockIdx.x * blockDim.x + threadIdx.x) << 2; g4 < nvox;
         g4 += (gridDim.x * blockDim.x) << 2) {
        const int b = g4 >> 18;
        const int v = g4 & (VOXPB - 1);
        const size_t pb = (size_t)b * 3 * VOXPB;   // plane base (x and D share layout)

        const int4   yv4 = *(const int4*)  (y + g4);
        const float4 d0v = *(const float4*)(D + pb + v);
        const float4 d1v = *(const float4*)(D + pb + VOXPB + v);
        const float4 d2v = *(const float4*)(D + pb + 2 * VOXPB + v);
        const float4 x1v = *(const float4*)(x + pb + VOXPB + v);
        const float4 x2v = *(const float4*)(x + pb + 2 * VOXPB + v);

        const int*   yv = (const int*)&yv4;
        const float* d0 = (const float*)&d0v;
        const float* d1 = (const float*)&d1v;
        const float* d2 = (const float*)&d2v;
        const float* x1 = (const float*)&x1v;
        const float* x2 = (const float*)&x2v;

        #pragma unroll
        for (int k = 0; k < 4; ++k) {
            const int yk = yv[k];
            const float m0 = (yk == 0) ? BIGF : d0[k];
            const float m1 = (yk == 1) ? BIGF : d1[k];
            const float m2 = (yk == 2) ? BIGF : d2[k];
            const float dt = sqrtf(fminf(m0, fminf(m1, m2))) + 1.0f;
            const float c1 = (yk == 1) ? (1.0f - x1[k]) : x1[k];
            const float c2 = (yk == 2) ? (1.0f - x2[k]) : x2[k];
            acc += dt * (c1 + c2);
        }
    }
    const float bs = block_sum_256(acc);   // all threads converged here
    if (threadIdx.x == 0) partials[blockIdx.x] = bs;
}

__global__ void final_kernel(const float* __restrict__ partials, int n,
                             float* __restrict__ out) {
    float acc = 0.0f;
    for (int i = threadIdx.x; i < n; i += blockDim.x) acc += partials[i];
    const float bs = block_sum_256(acc);
    if (threadIdx.x == 0)
        out[0] = bs * (1.0f / 1048576.0f) + 1e-5f;  // /N for mean(fp)+mean(fn), +SMOOTH
}

// ---------------------------------------------------------------------------
extern "C" void kernel_launch(void* const* d_in, const int* in_sizes, int n_in,
                              void* d_out, int out_size, void* d_ws, size_t ws_size,
                              hipStream_t stream) {
    const float* x = (const float*)d_in[0];   // (2,3,64,64,64) f32
    const int*   y = (const int*)d_in[1];     // (2,1,64,64,64) i32
    float* outp    = (float*)d_out;           // scalar f32

    float* bufA     = (float*)d_ws;                 // 6 * 262144 f32 (6 MB)
    float* bufB     = bufA + (size_t)NVOL * VOXPB;  // 6 MB
    float* partials = bufB + (size_t)NVOL * VOXPB;  // 512 f32

    const dim3 blk(256);
    const int lineBlocks = (NVOL * 4096) / 4; // 6144 blocks, 4 lines each

    // 3-axis separable quadratic min-plus EDT (init fused into z pass)
    minplus_pass<<<lineBlocks, blk, 0, stream>>>(nullptr, y, bufA, 2); // z
    minplus_pass<<<lineBlocks, blk, 0, stream>>>(bufA, nullptr, bufB, 1); // y
    minplus_pass<<<lineBlocks, blk, 0, stream>>>(bufB, nullptr, bufA, 0); // x

    // fused loss + deterministic reduction
    reduce_kernel<<<512, blk, 0, stream>>>(x, y, bufA, partials);
    final_kernel<<<1, blk, 0, stream>>>(partials, 512, outp);
}